// SingleLayerDecoder_386547057345
// MI455X (gfx1250) — compile-verified
//
#include <hip/hip_runtime.h>
#include <cstdint>

// Problem dims (fixed by the reference): B=16, L=512, E=128, H=64
#define DB 16
#define DL 512
#define DE 128
#define DH 64
#define DM (DB * DL)          // 8192 rows
#define NEG_BIG 1.0e8f
#define LOG2E 1.4426950408889634f
#define LN2   0.6931471805599453f

typedef float v2f __attribute__((ext_vector_type(2)));
typedef float v8f __attribute__((ext_vector_type(8)));

__device__ __forceinline__ float fast_exp(float x) {
  return __builtin_amdgcn_exp2f(x * LOG2E);        // v_exp_f32
}

__device__ __forceinline__ float fast_log1p01(float e) {
  // log1p(e) for e in (0, 1]: 1+e in (1,2], native v_log_f32 (base 2) * ln2.
  return LN2 * __builtin_amdgcn_logf(1.0f + e);
}

__device__ __forceinline__ float fast_tanh(float x) {
#if __has_builtin(__builtin_amdgcn_tanhf)
  return __builtin_amdgcn_tanhf(x);                // v_tanh_f32 on gfx1250
#else
  float ax = __builtin_fabsf(x);
  float t  = __builtin_amdgcn_exp2f(ax * (2.0f * LOG2E));
  float r  = 1.0f - 2.0f * __builtin_amdgcn_rcpf(t + 1.0f);
  return __builtin_copysignf(r, x);
#endif
}

// ---------------------------------------------------------------------------
// Kernel 1: fused projection GEMM via fp32 WMMA (V_WMMA_F32_16X16X4_F32).
// C = enc[8192x128] @ [W_l | W_r][128x128] -> dot_l[8192x64], dot_r[8192x64]
// One wave per 16x16 C tile; 8 waves per block cover one 16-row strip.
// A layout (16x4 f32): lanes 0-15 hold row M=lane, K={k,k+1};
//                      lanes 16-31 hold row M=lane-16, K={k+2,k+3}.
// B layout (4x16 f32): mirrored — lane holds its column, same K split.
// ---------------------------------------------------------------------------
__global__ __launch_bounds__(256) void proj_gemm_wmma(
    const float* __restrict__ enc, const float* __restrict__ Wl,
    const float* __restrict__ Wr, float* __restrict__ dotl,
    float* __restrict__ dotr) {
  const int wave  = threadIdx.x >> 5;   // 0..7 : n-tile (0-3 -> W_l, 4-7 -> W_r)
  const int lane  = threadIdx.x & 31;
  const int half  = lane >> 4;          // K sub-pair select
  const int l15   = lane & 15;
  const int mbase = blockIdx.x * 16;
  const int col   = ((wave & 3) << 4) + l15;

  const float* __restrict__ Wsel = (wave < 4) ? Wl : Wr;
  const float* ap = enc  + (size_t)(mbase + l15) * DE + half * 2;
  const float* bp = Wsel + (size_t)(half * 2) * DH + col;

  v8f c = {};
#pragma unroll 4
  for (int k = 0; k < DE; k += 4) {
    v2f a = *(const v2f*)ap;            // K = k+2*half, k+2*half+1 for row l15
    v2f b;
    b.x = bp[0];                        // W[k+2*half    ][col]
    b.y = bp[DH];                       // W[k+2*half + 1][col]
    c = __builtin_amdgcn_wmma_f32_16x16x4_f32(
        /*neg_a=*/false, a, /*neg_b=*/false, b,
        /*c_mod=*/(short)0, c, /*reuse_a=*/false, /*reuse_b=*/false);
    ap += 4;
    bp += 4 * DH;
  }

  float* __restrict__ dst = (wave < 4) ? dotl : dotr;
#pragma unroll
  for (int v = 0; v < 8; ++v) {
    const int row = mbase + v + half * 8;   // C layout: VGPR v, lane half
    dst[(size_t)row * DH + col] = c[v];
  }
}

// ---------------------------------------------------------------------------
// Kernel 2: pairwise biaffine scores + sigmoid/sample/entropy.
// One block per (b, i) row; dot_l[i,:] and U staged in LDS; each thread
// sweeps j. 64 v_tanh_f32 + FMA per output element (TRANS-bound, dot_r is
// L2-resident at 128 KB/batch).
// ---------------------------------------------------------------------------
__global__ __launch_bounds__(256) void pairwise_kernel(
    const float* __restrict__ dotl, const float* __restrict__ dotr,
    const float* __restrict__ U, const float* __restrict__ bias,
    float* __restrict__ out) {
  __shared__ float sdl[DH];
  __shared__ float su[DH];

  const int r = blockIdx.x;       // 0..8191
  const int b = r >> 9;           // r / 512
  const int i = r & (DL - 1);     // r % 512

  if (threadIdx.x < DH) {
    sdl[threadIdx.x] = dotl[(size_t)r * DH + threadIdx.x];
    su[threadIdx.x]  = U[threadIdx.x];
  }
  const float lb = bias[0];
  __syncthreads();

  const size_t N = (size_t)DB * DL * DL;   // 4,194,304 per output plane

  for (int j = threadIdx.x; j < DL; j += 256) {
    const float* drp = dotr + ((size_t)(b * DL + j)) * DH;
    float acc = 0.0f;
#pragma unroll
    for (int h = 0; h < DH; h += 4) {
      float4 d = *(const float4*)(drp + h);
      acc = __builtin_fmaf(su[h + 0], fast_tanh(sdl[h + 0] + d.x), acc);
      acc = __builtin_fmaf(su[h + 1], fast_tanh(sdl[h + 1] + d.y), acc);
      acc = __builtin_fmaf(su[h + 2], fast_tanh(sdl[h + 2] + d.z), acc);
      acc = __builtin_fmaf(su[h + 3], fast_tanh(sdl[h + 3] + d.w), acc);
    }
    float x = acc + lb;
    if (j == i) x -= NEG_BIG;               // diagonal (self-edge) mask

    const float p = __builtin_amdgcn_rcpf(1.0f + fast_exp(-x));  // sigmoid

    // Deterministic counter-based hash -> uniform -> Bernoulli sample.
    uint32_t s = (uint32_t)((uint32_t)r * (uint32_t)DL + (uint32_t)j);
    s = s * 0x9E3779B9u + 0x85EBCA6Bu;
    s ^= s >> 16; s *= 0x7FEB352Du;
    s ^= s >> 15; s *= 0x846CA68Bu;
    s ^= s >> 16;
    const float u = (float)s * 2.3283064365386963e-10f;   // [0,1)
    const float sample = (u < p) ? 1.0f : 0.0f;

    // Stable Bernoulli entropy: p*softplus(-x) + (1-p)*softplus(x)
    const float ax     = __builtin_fabsf(x);
    const float lse    = fast_log1p01(fast_exp(-ax));     // log1p(e^{-|x|})
    const float sp_pos = __builtin_fmaxf(x, 0.0f) + lse;  // softplus(x)
    const float sp_neg = __builtin_fmaxf(-x, 0.0f) + lse; // softplus(-x)
    const float entropy = p * sp_neg + (1.0f - p) * sp_pos;

    const size_t base = (size_t)r * DL + j;
    out[base]         = sample;   // plane 0: samples
    out[N + base]     = x;        // plane 1: masked scores
    out[2 * N + base] = entropy;  // plane 2: entropy
  }
}

extern "C" void kernel_launch(void* const* d_in, const int* in_sizes, int n_in,
                              void* d_out, int out_size, void* d_ws,
                              size_t ws_size, hipStream_t stream) {
  const float* enc  = (const float*)d_in[0];  // [16, 512, 128]
  const float* Wl   = (const float*)d_in[1];  // [128, 64]
  const float* Wr   = (const float*)d_in[2];  // [128, 64]
  const float* U    = (const float*)d_in[3];  // [64]
  const float* bias = (const float*)d_in[4];  // [1]

  float* ws   = (float*)d_ws;
  float* dotl = ws;                               // [8192, 64]
  float* dotr = ws + (size_t)DM * DH;             // [8192, 64]  (4 MB total)

  proj_gemm_wmma<<<DM / 16, 256, 0, stream>>>(enc, Wl, Wr, dotl, dotr);
  pairwise_kernel<<<DM, 256, 0, stream>>>(dotl, dotr, U, bias, (float*)d_out);
}